// Shift_gcn_3745211482515
// MI455X (gfx1250) — compile-verified
//
#include <hip/hip_runtime.h>
#include <math.h>

// CDNA5 / gfx1250, wave32. f32 WMMA (16x16x4) GEMM core — the network is
// memory-bound at f32 (~75 GFLOP vs ~2GB HBM traffic @ 23.3TB/s), so we keep
// full f32 precision and still ride the matrix pipe via V_WMMA_F32_16X16X4_F32.
// All GEMM mode flags are template parameters so the unrolled K-loop is
// straight-line: no per-iteration branches, no integer division, loads can be
// clause-batched and overlapped with the WMMA chain.

typedef float v2f __attribute__((ext_vector_type(2)));
typedef float v8f __attribute__((ext_vector_type(8)));

#define THREADS 256

constexpr int  kN    = 128;
constexpr int  kCin  = 64;
constexpr int  kT    = 128;
constexpr int  kV    = 19;
constexpr int  kCout = 128;
constexpr int  kCmid = 64;
constexpr int  kDG   = 32;
constexpr long kNT   = (long)kN * kT;   // 16384
constexpr long kR    = kNT * kV;        // 311296 (multiple of 16)
constexpr float kEps = 1e-5f;

// ---------------------------------------------------------------------------
// Generic f32 WMMA GEMM:  C[R,NC] = op(A)[R,K] @ B[K,NC] (+bias) (+C if ACC)
//   WKN=1 : B[k,n] = W[k*NC+n]   (row-major K x N, e.g. b*_lw)
//   WKN=0 : B[k,n] = W[n*K+k]    (conv-style [d,c] weights)
//   SHIFT=1: A element (row=(nt,v), ch) gathered from ntvc buffer via
//            shift-in index (v*K + ch*(K+1)) mod (K*V), times mask[v*K+ch].
//            The mod is strength-reduced to add + conditional subtract.
// A 16x4 f32 frag: lanes 0-15 = rows, vgpr{0,1} = K{0,1} (lanes 16-31: K{2,3}).
// B 4x16 f32 frag: lanes 0-15 = cols, vgpr{0,1} = K{0,1} (lanes 16-31: K{2,3}).
// C/D 16x16: vgpr r -> (M = r + 8*(lane>=16), N = lane&15).
// ---------------------------------------------------------------------------
template<int K, int NC, int SHIFT, int WKN, int ACC>
__global__ __launch_bounds__(THREADS) void wmma_gemm_f32(
    const float* __restrict__ A, const float* __restrict__ W,
    const float* __restrict__ bias, const float* __restrict__ mask,
    float* __restrict__ C, long R)
{
  constexpr int tilesN = NC / 16;            // 2, 4 or 8 -> division is a shift
  const int wave = threadIdx.x >> 5;
  const int lane = threadIdx.x & 31;
  const long tile  = (long)blockIdx.x * 8 + wave;
  const long tileM = tile / tilesN;
  const int  tileN = (int)(tile - tileM * tilesN);
  if (tileM * 16 >= R) return;               // wave-uniform exit (EXEC stays all-1s)

  const int  mlo  = lane & 15;
  const int  kb   = (lane >> 4) << 1;        // 0 or 2
  const long rowA = tileM * 16 + mlo;        // A-frag row for this lane
  const int  nB   = tileN * 16 + mlo;        // B-frag col / C col for this lane
  const int  mhi  = (lane >> 4) << 3;        // C-frag row group

  v8f acc;
  if (ACC) {
#pragma unroll
    for (int r = 0; r < 8; ++r)
      acc[r] = C[(tileM * 16 + mhi + r) * (long)NC + nB];
  } else {
#pragma unroll
    for (int r = 0; r < 8; ++r) acc[r] = 0.0f;
  }

  constexpr int KV = K * kV;
  const float* Aptr;
  const float* Mptr = nullptr;
  int idx0 = 0;
  if (SHIFT) {
    long nt  = rowA / kV;
    int vIdx = (int)(rowA - nt * kV);
    Aptr = A + nt * (long)KV;
    idx0 = vIdx * K + kb * (K + 1);
    if (idx0 >= KV) idx0 -= KV;              // max initial value < 2*KV
    Mptr = mask + vIdx * K + kb;
  } else {
    Aptr = A + rowA * (long)K + kb;
  }
  const float* Wptr = WKN ? (W + (long)kb * NC + nB)
                          : (W + (long)nB * K + kb);

#pragma unroll
  for (int k0 = 0; k0 < K; k0 += 4) {
    v2f a, b;
    if (SHIFT) {
      int idx1 = idx0 + (K + 1); if (idx1 >= KV) idx1 -= KV;
      a.x = Aptr[idx0] * Mptr[0];
      a.y = Aptr[idx1] * Mptr[1];
      idx0 += 4 * (K + 1); if (idx0 >= KV) idx0 -= KV;  // 4*(K+1) < KV
      Mptr += 4;
    } else {
      a.x = Aptr[0]; a.y = Aptr[1];
      Aptr += 4;
    }
    if (WKN) { b.x = Wptr[0]; b.y = Wptr[NC]; Wptr += 4 * NC; }
    else     { b.x = Wptr[0]; b.y = Wptr[1];  Wptr += 4; }
    acc = __builtin_amdgcn_wmma_f32_16x16x4_f32(
        false, a, false, b, (short)0, acc, false, false);
  }

  const float bv = bias ? bias[nB] : 0.0f;
#pragma unroll
  for (int r = 0; r < 8; ++r)
    C[(tileM * 16 + mhi + r) * (long)NC + nB] = acc[r] + bv;
}

// ---------------------------------------------------------------------------
// Column-wise sum / sum-of-squares reduction (for train-mode BN statistics).
// ---------------------------------------------------------------------------
__global__ __launch_bounds__(THREADS) void zero_f32(float* __restrict__ p, long n)
{
  long i = (long)blockIdx.x * THREADS + threadIdx.x;
  if (i < n) p[i] = 0.0f;
}

__global__ __launch_bounds__(THREADS) void col_reduce(
    const float* __restrict__ X, long nrows, int ncols, long chunk,
    float* __restrict__ sum, float* __restrict__ sumsq)
{
  int col = blockIdx.x * THREADS + threadIdx.x;
  if (col >= ncols) return;
  long r0 = (long)blockIdx.y * chunk;
  long r1 = r0 + chunk; if (r1 > nrows) r1 = nrows;
  float s = 0.0f, ss = 0.0f;
  for (long r = r0; r < r1; ++r) {
    float v = X[r * (long)ncols + col];
    s += v; ss += v * v;
  }
  atomicAdd(&sum[col], s);
  atomicAdd(&sumsq[col], ss);
}

// Build per-feature affine (scale, bias). If permute_so, feature p reads the
// statistics of source column so[p] = (v*cout + d*(1-cout)) mod (cout*V).
__global__ __launch_bounds__(THREADS) void bn_finalize(
    const float* __restrict__ sum, const float* __restrict__ sumsq,
    const float* __restrict__ gamma, const float* __restrict__ beta,
    float* __restrict__ scale, float* __restrict__ bias,
    long nrows, int ncols, int permute_so, int cout)
{
  int p = blockIdx.x * THREADS + threadIdx.x;
  if (p >= ncols) return;
  int src = p;
  if (permute_so) {
    int v = p / cout, d = p - v * cout;
    long M = (long)cout * kV;
    long t = ((long)v * cout + (long)d * (1 - cout)) % M;
    if (t < 0) t += M;
    src = (int)t;
  }
  float inv = 1.0f / (float)nrows;
  float m = sum[src] * inv;
  float var = sumsq[src] * inv - m * m;
  float sc = gamma[p] * rsqrtf(var + kEps);
  scale[p] = sc;
  bias[p]  = beta[p] - m * sc;
}

// x2 = relu( BN1d(shift_out(xr)) + residual ), output in ntvc layout.
__global__ __launch_bounds__(THREADS) void apply_x2(
    const float* __restrict__ xr, const float* __restrict__ scale,
    const float* __restrict__ bias, const float* __restrict__ resid,
    const float* __restrict__ rscale, const float* __restrict__ rbias,
    float* __restrict__ x2, int cout)
{
  long i = (long)blockIdx.x * THREADS + threadIdx.x;
  long total = kR * (long)cout;
  if (i >= total) return;
  int d = (int)(i % cout);
  long rv = i / cout;
  int v = (int)(rv % kV);
  long nt = rv / kV;
  int p = v * cout + d;
  long M = (long)cout * kV;
  long t = ((long)v * cout + (long)d * (1 - cout)) % M;
  if (t < 0) t += M;
  float xv = xr[nt * M + t] * scale[p] + bias[p];
  float rr = 0.0f;
  if (resid) {
    rr = resid[i];
    if (rscale) rr = rr * rscale[d] + rbias[d];
  }
  x2[i] = fmaxf(xv + rr, 0.0f);
}

// Per-(n,t): scores = A[19,32] @ B[19,32]^T, row-softmax -> g[nt,19,19]. LDS staged.
__global__ __launch_bounds__(THREADS) void g_softmax(
    const float* __restrict__ abuf, const float* __restrict__ bbuf,
    float* __restrict__ g)
{
  __shared__ float sa[kV * kDG];
  __shared__ float sb[kV * kDG];
  __shared__ float sc[kV * kV];
  long nt = blockIdx.x;
  const float* ap = abuf + nt * (long)kV * kDG;
  const float* bp = bbuf + nt * (long)kV * kDG;
  for (int i = threadIdx.x; i < kV * kDG; i += THREADS) { sa[i] = ap[i]; sb[i] = bp[i]; }
  __syncthreads();
  for (int i = threadIdx.x; i < kV * kV; i += THREADS) {
    int u = i / kV, w = i - u * kV;
    float s = 0.0f;
    for (int k = 0; k < kDG; ++k) s += sa[u * kDG + k] * sb[w * kDG + k];
    sc[i] = s;
  }
  __syncthreads();
  float* gp = g + nt * (long)kV * kV;
  for (int u = threadIdx.x; u < kV; u += THREADS) {
    float mx = -1e30f;
    for (int w = 0; w < kV; ++w) mx = fmaxf(mx, sc[u * kV + w]);
    float se = 0.0f;
    for (int w = 0; w < kV; ++w) se += expf(sc[u * kV + w] - mx);
    float inv = 1.0f / se;
    for (int w = 0; w < kV; ++w) gp[u * kV + w] = expf(sc[u * kV + w] - mx) * inv;
  }
}

// Per-(n,t): xg[19,cout] = g[19,19] @ x2[19,cout]. g staged in LDS.
__global__ __launch_bounds__(THREADS) void attn_apply(
    const float* __restrict__ g, const float* __restrict__ x2,
    float* __restrict__ xg, int cout)
{
  __shared__ float sg[kV * kV];
  long nt = blockIdx.x;
  for (int i = threadIdx.x; i < kV * kV; i += THREADS) sg[i] = g[nt * (long)kV * kV + i];
  __syncthreads();
  const float* xp = x2 + nt * (long)kV * cout;
  float* op = xg + nt * (long)kV * cout;
  for (int i = threadIdx.x; i < kV * cout; i += THREADS) {
    int u = i / cout, d = i - u * cout;
    float s = 0.0f;
    for (int w = 0; w < kV; ++w) s += sg[u * kV + w] * xp[w * cout + d];
    op[i] = s;
  }
}

// [N,C,T,V] -> [(n,t), v, c]  (ntvc internal layout)
__global__ __launch_bounds__(THREADS) void nchw_to_ntvc(
    const float* __restrict__ x, float* __restrict__ y, int C)
{
  long i = (long)blockIdx.x * THREADS + threadIdx.x;
  long total = (long)kN * C * kT * kV;
  if (i >= total) return;
  int c = (int)(i % C);
  long rv = i / C;
  int v = (int)(rv % kV);
  long nt = rv / kV;
  long n = nt / kT; int t = (int)(nt - n * kT);
  y[i] = x[((n * C + c) * (long)kT + t) * kV + v];
}

// out = relu(BN2d(y)); optionally transposed back to [n,c,t,v] (final output).
__global__ __launch_bounds__(THREADS) void apply_y(
    const float* __restrict__ y, const float* __restrict__ scale,
    const float* __restrict__ bias, float* __restrict__ out,
    int cout, int to_nchw)
{
  long i = (long)blockIdx.x * THREADS + threadIdx.x;
  long total = kR * (long)cout;
  if (i >= total) return;
  int d = (int)(i % cout);
  long rv = i / cout;
  int v = (int)(rv % kV);
  long nt = rv / kV;
  float val = fmaxf(y[i] * scale[d] + bias[d], 0.0f);
  if (to_nchw) {
    long n = nt / kT; int t = (int)(nt - n * kT);
    out[((n * cout + d) * (long)kT + t) * kV + v] = val;
  } else {
    out[i] = val;
  }
}

__global__ __launch_bounds__(THREADS) void make_mask(
    const float* __restrict__ fm, float* __restrict__ mask, int n)
{
  int i = blockIdx.x * THREADS + threadIdx.x;
  if (i < n) mask[i] = tanhf(fm[i]) + 1.0f;
}

// ---------------------------------------------------------------------------
// Host-side launch helpers
// ---------------------------------------------------------------------------
static void launch_gemm(const float* A, const float* W, const float* bias,
                        const float* mask, float* C, long R, int K, int NC,
                        int shift, int accum, hipStream_t s)
{
  long tiles = (R / 16) * (long)(NC / 16);
  int blocks = (int)((tiles + 7) / 8);
#define GEMM_CASE(KK, NN, SS, WW, AA)                                          \
  wmma_gemm_f32<KK, NN, SS, WW, AA><<<blocks, THREADS, 0, s>>>(A, W, bias,     \
                                                              mask, C, R)
  if (K == 64 && NC == 32) { GEMM_CASE(64, 32, 0, 0, 0); return; }
  if (K == 64 && NC == 64) {
    if (shift) { GEMM_CASE(64, 64, 1, 1, 0); return; }
    if (accum) { GEMM_CASE(64, 64, 0, 0, 1); return; }
    GEMM_CASE(64, 64, 0, 0, 0); return;
  }
  if (K == 64 && NC == 128) {
    if (shift) { GEMM_CASE(64, 128, 1, 1, 0); return; }
    GEMM_CASE(64, 128, 0, 0, 0); return;
  }
  // K == 128, NC == 128
  if (shift) { GEMM_CASE(128, 128, 1, 1, 0); return; }
  if (accum) { GEMM_CASE(128, 128, 0, 0, 1); return; }
  GEMM_CASE(128, 128, 0, 0, 0);
#undef GEMM_CASE
}

static void launch_reduce(const float* X, long nrows, int ncols,
                          float* sum, float* ssq, hipStream_t s)
{
  int zb = (ncols + THREADS - 1) / THREADS;
  zero_f32<<<zb, THREADS, 0, s>>>(sum, ncols);
  zero_f32<<<zb, THREADS, 0, s>>>(ssq, ncols);
  long chunk = 1024;
  dim3 grid((unsigned)((ncols + THREADS - 1) / THREADS),
            (unsigned)((nrows + chunk - 1) / chunk));
  col_reduce<<<grid, THREADS, 0, s>>>(X, nrows, ncols, chunk, sum, ssq);
}

extern "C" void kernel_launch(void* const* d_in, const int* in_sizes, int n_in,
                              void* d_out, int out_size, void* d_ws, size_t ws_size,
                              hipStream_t stream)
{
  (void)in_sizes; (void)n_in; (void)out_size;
  auto F = [&](int i) { return (const float*)d_in[i]; };

  const long RC = kR * (long)kCout;          // 39,845,888 floats
  const long need = 5 * RC + kNT * (long)kV * kV + 32768;
  if (ws_size < (size_t)need * sizeof(float)) return;  // workspace too small

  float* ws = (float*)d_ws;
  float* XA = ws;                 // activation ping  [R, <=128] ntvc
  float* XB = XA + RC;            // activation pong
  float* T1 = XB + RC;            // xr / y / a_buf
  float* T2 = T1 + RC;            // x2
  float* T3 = T2 + RC;            // res0 / xg / b_buf
  float* G  = T3 + RC;            // attention [NT,19,19]
  float* SM = G + kNT * (long)kV * kV;
  float* ssum = SM;         float* ssq = SM + 4096;
  float* sc1  = SM + 8192;  float* bi1 = SM + 12288;
  float* sc2  = SM + 16384; float* bi2 = SM + 20480;
  float* msk  = SM + 24576;

  // ---- x0 -> ntvc ----
  {
    long tot = (long)kN * kCin * kT * kV;
    nchw_to_ntvc<<<(int)((tot + THREADS - 1) / THREADS), THREADS, 0, stream>>>(F(0), XA, kCin);
  }

  // ---- compute_g_spa: two projections + scores + softmax ----
  launch_gemm(XA, F(1), F(2), nullptr, T1, kR, kCin, kDG, 0, 0, stream); // a
  launch_gemm(XA, F(3), F(4), nullptr, T3, kR, kCin, kDG, 0, 0, stream); // b
  g_softmax<<<(int)kNT, THREADS, 0, stream>>>(T1, T3, G);

  // ---- one GCN block ----
  auto run_block = [&](const float* Xin, int cin, int cout,
                       const float* lw, const float* lb, const float* fm,
                       const float* bng, const float* bnb,
                       const float* bnsg, const float* bnsb,
                       const float* ww, const float* w1w, const float* w1b,
                       const float* dw, const float* db,
                       const float* dbng, const float* dbnb,
                       float* Xout, int to_nchw)
  {
    long total = kR * (long)cout;
    int ab = (int)((total + THREADS - 1) / THREADS);

    make_mask<<<(kV * cin + THREADS - 1) / THREADS, THREADS, 0, stream>>>(fm, msk, kV * cin);

    // xr = (shift_in(x)*mask) @ lw + lb     -> T1 [R,cout] == [NT, V*cout]
    launch_gemm(Xin, lw, lb, msk, T1, kR, cin, cout, 1, 0, stream);

    // BN1d stats on xr columns; affine indexed post shift-out permutation
    launch_reduce(T1, kNT, kV * cout, ssum, ssq, stream);
    bn_finalize<<<(kV * cout + THREADS - 1) / THREADS, THREADS, 0, stream>>>(
        ssum, ssq, bng, bnb, sc1, bi1, kNT, kV * cout, 1, cout);

    const float* resid = Xin;
    const float* rs = nullptr; const float* rb = nullptr;
    if (dw) {  // down path: res = BN2d(conv1x1(x))
      launch_gemm(Xin, dw, db, nullptr, T3, kR, cin, cout, 0, 0, stream);
      launch_reduce(T3, kR, cout, ssum, ssq, stream);
      bn_finalize<<<(cout + THREADS - 1) / THREADS, THREADS, 0, stream>>>(
          ssum, ssq, dbng, dbnb, sc2, bi2, kR, cout, 0, cout);
      resid = T3; rs = sc2; rb = bi2;
    }

    // x2 = relu(BN(shift_out(xr)) + res)    -> T2 [R,cout] ntvc
    apply_x2<<<ab, THREADS, 0, stream>>>(T1, sc1, bi1, resid, rs, rb, T2, cout);

    // xg = g @ x2 per (n,t)                 -> T3
    attn_apply<<<(int)kNT, THREADS, 0, stream>>>(G, T2, T3, cout);

    // y = xg @ ww^T + x2 @ w1w^T + w1b      -> T1 (second GEMM accumulates)
    launch_gemm(T3, ww, nullptr, nullptr, T1, kR, cout, cout, 0, 0, stream);
    launch_gemm(T2, w1w, w1b, nullptr, T1, kR, cout, cout, 0, 1, stream);

    // BN2d(y) + relu -> Xout
    launch_reduce(T1, kR, cout, ssum, ssq, stream);
    bn_finalize<<<(cout + THREADS - 1) / THREADS, THREADS, 0, stream>>>(
        ssum, ssq, bnsg, bnsb, sc1, bi1, kR, cout, 0, cout);
    apply_y<<<ab, THREADS, 0, stream>>>(T1, sc1, bi1, Xout, cout, to_nchw);
  };

  // block 1: 64 -> 64, identity residual
  run_block(XA, kCin, kCmid, F(5), F(6), F(7), F(8), F(9), F(10), F(11),
            F(12), F(13), F(14), nullptr, nullptr, nullptr, nullptr, XB, 0);
  // block 2: 64 -> 128, conv+BN down path
  run_block(XB, kCmid, kCout, F(15), F(16), F(17), F(18), F(19), F(20), F(21),
            F(22), F(23), F(24), F(25), F(26), F(27), F(28), XA, 0);
  // block 3: 128 -> 128, identity residual; emit final [n,c,t,v]
  run_block(XA, kCout, kCout, F(29), F(30), F(31), F(32), F(33), F(34), F(35),
            F(36), F(37), F(38), nullptr, nullptr, nullptr, nullptr,
            (float*)d_out, 1);
}